// RoiAlignTPU_78769700208950
// MI455X (gfx1250) — compile-verified
//
#include <hip/hip_runtime.h>
#include <hip/hip_bf16.h>

#define CROP 14
#define CH   256
#define EPSF 1e-07f

typedef float v4f __attribute__((ext_vector_type(4)));

// Grid = (49, N, B): each block owns 4 output pixels of ONE box.
// b, n come from blockIdx (SGPRs) => all box/level math scalarizes to SALU
// and the boxes read becomes a scalar load. Each 64-thread pixel group does
// 4 fully coalesced 1KB global_load_b128 tap bursts; output is written with
// non-temporal b128 stores so the 103MB result stream does not evict the
// ~45MB feature pyramid from the 192MB L2.
__global__ __launch_bounds__(256) void roi_align_gfx1250(
    const int*   __restrict__ image_shape,   // [4] = B,H,W,3
    const float* __restrict__ boxes,         // [B,N,4] x1,y1,x2,y2
    const float* __restrict__ f3,            // [B,128,128,C]
    const float* __restrict__ f4,            // [B, 64, 64,C]
    const float* __restrict__ f5,            // [B, 32, 32,C]
    const float* __restrict__ f6,            // [B, 16, 16,C]
    const float* __restrict__ f7,            // [B,  8,  8,C]
    float*       __restrict__ out,           // [B,N,14,14,C]
    int N)
{
    const int tid = threadIdx.x;
    const int n   = blockIdx.y;              // SGPR
    const int b   = blockIdx.z;              // SGPR

    const int pib = blockIdx.x * 4 + (tid >> 6);   // pixel-in-box, 0..195
    const int c0  = (tid & 63) * 4;                // channel offset (16B aligned)
    const int i   = pib / CROP;                    // constant divisor -> cheap
    const int j   = pib - i * CROP;

    // ---- box / level math: depends only on SGPR inputs -> scalarizes ----
    const float h1 = (float)image_shape[1] - 1.0f;       // 1023
    const float w1 = (float)image_shape[2] - 1.0f;       // 1023

    const v4f bx = *(const v4f*)(boxes + (size_t)(b * N + n) * 4);
    float x1 = fminf(fmaxf(bx.x, 0.0f), w1);
    float y1 = fminf(fmaxf(bx.y, 0.0f), h1);
    float x2 = fminf(fmaxf(bx.z, 0.0f), w1);
    float y2 = fminf(fmaxf(bx.w, 0.0f), h1);
    // second clip: clip(boxes, 0, [x2,y2,x2,y2])
    x1 = fminf(x1, x2);
    y1 = fminf(y1, y2);

    const float area_sqrt = sqrtf((x2 - x1) * (y2 - y1));
    float lvl = floorf(4.0f + log2f(area_sqrt * (1.0f / 224.0f) + EPSF));
    lvl = fminf(fmaxf(lvl, 3.0f), 7.0f);
    const int   fidx = (int)lvl - 3;                     // 0..4
    const int   s    = 128 >> fidx;                      // feature H==W
    const int   bnd  = s - 1;
    const float inv  = 1.0f / (float)(1 << (int)lvl);    // exact 2^-lvl

    const float ly1 = y1 * inv, lx1 = x1 * inv;
    const float sy  = (y2 * inv - ly1) / (float)CROP;    // per-bin step
    const float sx  = (x2 * inv - lx1) / (float)CROP;

    const float* f;
    switch (fidx) {
        case 0:  f = f3; break;
        case 1:  f = f4; break;
        case 2:  f = f5; break;
        case 3:  f = f6; break;
        default: f = f7; break;
    }

    // ---- per-pixel sample point + bilinear weights (vector, minimal) ----
    const float gy = ((float)i + 0.5f) * sy + ly1;
    const float gx = ((float)j + 0.5f) * sx + lx1;

    const int y0  = min(max((int)floorf(gy), 0), bnd);
    const int y1i = min(y0 + 1, bnd);
    const int x0  = min(max((int)floorf(gx), 0), bnd);
    const int x1i = min(x0 + 1, bnd);

    // fractional weights use the *clipped* integer corner (matches reference)
    const float fy = gy - (float)y0;
    const float fx = gx - (float)x0;
    const float w00 = (1.0f - fy) * (1.0f - fx);
    const float w01 = (1.0f - fy) * fx;
    const float w10 = fy * (1.0f - fx);
    const float w11 = fy * fx;

    // ---- gather 4 taps (each a coalesced 1KB burst per pixel group) ----
    const size_t base = (size_t)b * s * s * CH + c0;
    const v4f v00 = *(const v4f*)(f + base + (size_t)(y0  * s + x0 ) * CH);
    const v4f v01 = *(const v4f*)(f + base + (size_t)(y0  * s + x1i) * CH);
    const v4f v10 = *(const v4f*)(f + base + (size_t)(y1i * s + x0 ) * CH);
    const v4f v11 = *(const v4f*)(f + base + (size_t)(y1i * s + x1i) * CH);

    v4f r;
    r.x = fmaf(w00, v00.x, fmaf(w01, v01.x, fmaf(w10, v10.x, w11 * v11.x)));
    r.y = fmaf(w00, v00.y, fmaf(w01, v01.y, fmaf(w10, v10.y, w11 * v11.y)));
    r.z = fmaf(w00, v00.z, fmaf(w01, v01.z, fmaf(w10, v10.z, w11 * v11.z)));
    r.w = fmaf(w00, v00.w, fmaf(w01, v01.w, fmaf(w10, v10.w, w11 * v11.w)));

    // Non-temporal streaming store: output is write-once, keep L2 for features.
    const size_t opix = ((size_t)(b * N + n) * (CROP * CROP) + pib) * CH + c0;
    __builtin_nontemporal_store(r, (v4f*)(out + opix));
}

extern "C" void kernel_launch(void* const* d_in, const int* in_sizes, int n_in,
                              void* d_out, int out_size, void* d_ws, size_t ws_size,
                              hipStream_t stream) {
    const int*   image_shape = (const int*)  d_in[0];
    const float* boxes       = (const float*)d_in[1];
    // d_in[2] = bat_scores (unused by the reference output)
    const float* f3          = (const float*)d_in[3];
    const float* f4          = (const float*)d_in[4];
    const float* f5          = (const float*)d_in[5];
    const float* f6          = (const float*)d_in[6];
    const float* f7          = (const float*)d_in[7];
    float*       out         = (float*)d_out;

    const int BN = in_sizes[2];                       // B*N (scores are [B,N])
    const int B  = in_sizes[3] / (128 * 128 * CH);    // from feat3 element count
    const int N  = BN / B;

    // 196 pixels per box, 4 pixels per 256-thread block -> 49 blocks in x.
    dim3 grid(49, N, B);
    roi_align_gfx1250<<<grid, 256, 0, stream>>>(
        image_shape, boxes, f3, f4, f5, f6, f7, out, N);
}